// gcn_23098334118566
// MI455X (gfx1250) — compile-verified
//
#include <hip/hip_runtime.h>

#define DIM 64

typedef __attribute__((ext_vector_type(2))) float v2f;
typedef __attribute__((ext_vector_type(4))) float v4f;
typedef __attribute__((ext_vector_type(8))) float v8f;

// ---------------------------------------------------------------------------
// init: acc = cur = concat(emb_user, emb_item), float4-vectorized stream copy
// ---------------------------------------------------------------------------
__global__ void gcn_init_concat(const float* __restrict__ eu,
                                const float* __restrict__ ei,
                                float* __restrict__ cur,
                                float* __restrict__ acc,
                                long userVec4, long totalVec4) {
    long i = (long)blockIdx.x * blockDim.x + threadIdx.x;
    if (i >= totalVec4) return;
    v4f v = (i < userVec4) ? ((const v4f*)eu)[i] : ((const v4f*)ei)[i - userVec4];
    ((v4f*)cur)[i] = v;
    ((v4f*)acc)[i] = v;
}

// ---------------------------------------------------------------------------
// zero: p = 0 (float4 stream) -- used once, before the first layer
// ---------------------------------------------------------------------------
__global__ void gcn_zero(float* __restrict__ p, long totalVec4) {
    long i = (long)blockIdx.x * blockDim.x + threadIdx.x;
    if (i >= totalVec4) return;
    v4f z = {0.f, 0.f, 0.f, 0.f};
    ((v4f*)p)[i] = z;
}

// ---------------------------------------------------------------------------
// SpMM edge kernel: 16 threads per edge, each owns one float4 chunk of the
// 64-float row.  Gather of cur[col] is a fully coalesced 256B read per edge
// and stays L2-resident (node buffers ~115MB < 192MB L2).  Edge metadata
// (rows/cols/vals, 57.6MB streamed once per layer) is loaded non-temporally
// so it does not evict the node working set from L2.  Scatter is 4x native
// f32 atomic-add, no-return form (relaxed, agent scope -> STOREcnt path).
// ---------------------------------------------------------------------------
__global__ void gcn_spmm(const int* __restrict__ rows,
                         const int* __restrict__ cols,
                         const float* __restrict__ vals,
                         const float* __restrict__ cur,
                         float* __restrict__ nxt,
                         long nEdges) {
    long t = (long)blockIdx.x * blockDim.x + threadIdx.x;
    long e = t >> 4;
    if (e >= nEdges) return;
    int c4 = (int)(t & 15);
    int row = __builtin_nontemporal_load(rows + e);
    int col = __builtin_nontemporal_load(cols + e);
    float v = __builtin_nontemporal_load(vals + e);
    v4f x = ((const v4f*)(cur + (long)col * DIM))[c4];
    float* dst = nxt + (long)row * DIM + c4 * 4;
    (void)__hip_atomic_fetch_add(dst + 0, v * x.x, __ATOMIC_RELAXED, __HIP_MEMORY_SCOPE_AGENT);
    (void)__hip_atomic_fetch_add(dst + 1, v * x.y, __ATOMIC_RELAXED, __HIP_MEMORY_SCOPE_AGENT);
    (void)__hip_atomic_fetch_add(dst + 2, v * x.z, __ATOMIC_RELAXED, __HIP_MEMORY_SCOPE_AGENT);
    (void)__hip_atomic_fetch_add(dst + 3, v * x.w, __ATOMIC_RELAXED, __HIP_MEMORY_SCOPE_AGENT);
}

// ---------------------------------------------------------------------------
// fused: acc += nxt, and zero the retired cur buffer (it becomes the next
// layer's scatter destination).  Saves one full streaming pass per layer.
// ---------------------------------------------------------------------------
__global__ void gcn_acc_add_zero(float* __restrict__ acc,
                                 const float* __restrict__ nxt,
                                 float* __restrict__ retired,
                                 long totalVec4) {
    long i = (long)blockIdx.x * blockDim.x + threadIdx.x;
    if (i >= totalVec4) return;
    v4f a = ((const v4f*)acc)[i];
    v4f b = ((const v4f*)nxt)[i];
    a.x += b.x; a.y += b.y; a.z += b.z; a.w += b.w;
    ((v4f*)acc)[i] = a;
    v4f z = {0.f, 0.f, 0.f, 0.f};
    ((v4f*)retired)[i] = z;
}

// ---------------------------------------------------------------------------
// Final stage: out[b] = dot(acc[users[b]], acc[N_USERS+items[b']]) over D=64, b' = b%16<8 ? b : b-8.
// One wave (32 lanes) computes 16 dots as entries of U * I^T with a chain of 16
// V_WMMA_F32_16X16X4_F32 ops (full FP32 matrix pipe, exact in f32).
//
// A-frag layout (16x4 f32): lanes 0-15 hold M=lane, K=kb+{0,1}; lanes 16-31
// hold M=lane-16, K=kb+{2,3}.  B-frag (4x16) mirrors with N in place of M.
// C/D layout: VGPR r, lanes 0-15 -> (M=r, N=lane); lanes 16-31 -> (M=8+r).
// Index clamping (no branch) keeps EXEC all-ones as WMMA requires.
// ---------------------------------------------------------------------------
__global__ void gcn_dot_wmma(const float* __restrict__ acc,
                             const int* __restrict__ users,
                             const int* __restrict__ items,
                             float* __restrict__ out,
                             int nUser, int batch) {
    __shared__ float tile[32 * 8];
    int lane = threadIdx.x;           // 0..31, one wave
    int b0 = blockIdx.x * 16;
    int m = lane & 15;
    int half = lane >> 4;             // which K-pair this lane holds
    int b = b0 + m;
    if (b >= batch) b = batch - 1;    // clamp, no divergence
    const float* U = acc + (long)users[b] * DIM;
    const float* I = acc + ((long)nUser + items[b]) * DIM;

    v8f c = {};
#pragma unroll
    for (int kk = 0; kk < 16; ++kk) {
        int k = kk * 4 + half * 2;
        v2f a;  a.x = U[k];  a.y = U[k + 1];
        v2f bb; bb.x = I[k]; bb.y = I[k + 1];
        c = __builtin_amdgcn_wmma_f32_16x16x4_f32(
                /*neg_a=*/false, a, /*neg_b=*/false, bb,
                /*c_mod=*/(short)0, c, /*reuse_a=*/false, /*reuse_b=*/false);
    }

#pragma unroll
    for (int r = 0; r < 8; ++r) tile[lane * 8 + r] = c[r];
    __syncthreads();

    if (lane < 16 && (b0 + lane) < batch) {
        int mm = lane;
        int srcLane = (mm < 8) ? mm : mm + 8;   // lane holding (mm, mm) if mm<8, else (mm, mm-8)
        int srcReg  = (mm < 8) ? mm : mm - 8;   // VGPR index holding it
        out[b0 + mm] = tile[srcLane * 8 + srcReg];
    }
}

// ---------------------------------------------------------------------------
extern "C" void kernel_launch(void* const* d_in, const int* in_sizes, int n_in,
                              void* d_out, int out_size, void* d_ws, size_t ws_size,
                              hipStream_t stream) {
    const float* eu   = (const float*)d_in[0];
    const float* ei   = (const float*)d_in[1];
    const int*   rows = (const int*)d_in[2];
    const int*   cols = (const int*)d_in[3];
    const float* vals = (const float*)d_in[4];
    const int*   usr  = (const int*)d_in[5];
    const int*   itm  = (const int*)d_in[6];
    float* out = (float*)d_out;

    int  nUser  = in_sizes[0] / DIM;
    int  nItem  = in_sizes[1] / DIM;
    int  nNode  = nUser + nItem;
    long nEdges = in_sizes[2];
    int  batch  = in_sizes[5];

    long nodeFloats = (long)nNode * DIM;
    long nVec4      = nodeFloats / 4;
    long userVec4   = (long)nUser * DIM / 4;

    float* bufA = (float*)d_ws;             // cur
    float* bufB = bufA + nodeFloats;        // nxt
    float* accb = bufB + nodeFloats;        // acc (layer sum)

    const int T = 256;
    dim3 gNode((unsigned)((nVec4 + T - 1) / T));

    gcn_init_concat<<<gNode, T, 0, stream>>>(eu, ei, bufA, accb, userVec4, nVec4);
    gcn_zero<<<gNode, T, 0, stream>>>(bufB, nVec4);

    float* cur = bufA;
    float* nxt = bufB;
    long edgeThreads = nEdges * 16;
    dim3 gEdge((unsigned)((edgeThreads + T - 1) / T));

    for (int layer = 0; layer < 3; ++layer) {
        gcn_spmm<<<gEdge, T, 0, stream>>>(rows, cols, vals, cur, nxt, nEdges);
        // acc += nxt; zero retired cur (it is the next layer's scatter target)
        gcn_acc_add_zero<<<gNode, T, 0, stream>>>(accb, nxt, cur, nVec4);
        float* tmp = cur; cur = nxt; nxt = tmp;
    }

    dim3 gDot((unsigned)((batch + 15) / 16));
    gcn_dot_wmma<<<gDot, 32, 0, stream>>>(accb, usr, itm, out, nUser, batch);
}